// GlobalNode_6133213299117
// MI455X (gfx1250) — compile-verified
//
#include <hip/hip_runtime.h>
#include <hip/hip_bf16.h>
#include <math.h>

typedef __attribute__((ext_vector_type(2))) float v2f;
typedef __attribute__((ext_vector_type(8))) float v8f;

#define D 256
#define SLOPE 0.01f

__device__ __forceinline__ v8f wmma_f32(v2f a, v2f b, v8f c) {
    // D = A(16x4) * B(4x16) + C, fp32, wave32
    return __builtin_amdgcn_wmma_f32_16x16x4_f32(
        /*neg_a=*/false, a, /*neg_b=*/false, b,
        /*c_mod=*/(short)0, c, /*reuse_a=*/false, /*reuse_b=*/false);
}

__device__ __forceinline__ float leaky(float v) {
    return v >= 0.f ? v : SLOPE * v;
}

// ---------------------------------------------------------------- init
__global__ void init_kernel(float* gmax, float* gsum, float* xg, int B, int BD) {
    int i = blockIdx.x * blockDim.x + threadIdx.x;
    if (i < B) { gmax[i] = -__builtin_inff(); gsum[i] = 0.f; }
    if (i < BD) xg[i] = 0.f;
}

// --------------------------------------------- gate GEMV + segment max
// one wave per node: lane l covers cols l, l+32, ..., l+224
__global__ __launch_bounds__(256)
void gate_max_kernel(const float* __restrict__ x, const int* __restrict__ batch,
                     const float* __restrict__ Wg, const float* __restrict__ bg,
                     float* __restrict__ gate, float* __restrict__ gmax, int N) {
    int wave = threadIdx.x >> 5;
    int lane = threadIdx.x & 31;
    int node = blockIdx.x * 8 + wave;
    if (node >= N) return;
    const float* xr = x + (size_t)node * D;
    float acc = 0.f;
#pragma unroll
    for (int i = 0; i < 8; ++i) {
        int c = lane + i * 32;
        acc += xr[c] * Wg[c];
    }
#pragma unroll
    for (int off = 16; off > 0; off >>= 1)
        acc += __shfl_xor(acc, off, 32);
    if (lane == 0) {
        float g = acc + bg[0];
        gate[node] = g;
        int b = batch[node];
        // sign-aware float max via integer atomics (init = -inf)
        if (g >= 0.f) atomicMax((int*)(gmax + b), __float_as_int(g));
        else          atomicMin((unsigned int*)(gmax + b),
                                (unsigned int)__float_as_int(g));
    }
}

// ------------------------------------------------- segment sum of exp
__global__ void expsum_kernel(const float* __restrict__ gate,
                              const int* __restrict__ batch,
                              const float* __restrict__ gmax,
                              float* __restrict__ gsum, int N) {
    int i = blockIdx.x * blockDim.x + threadIdx.x;
    if (i < N) {
        int b = batch[i];
        atomicAdd(&gsum[b], __expf(gate[i] - gmax[b]));
    }
}

// ------------- feat GEMM (WMMA) + LeakyReLU + alpha-weighted segment sum
// block: 64 nodes x 256 cols; 8 waves, each wave owns 32 cols (2 C tiles),
// iterates 4 M-subtiles reusing B fragments. A tile in 64KB LDS, XOR-swizzled.
__global__ __launch_bounds__(256)
void feat_agg_kernel(const float* __restrict__ x, const int* __restrict__ batch,
                     const float* __restrict__ Wf, const float* __restrict__ bf,
                     const float* __restrict__ gate, const float* __restrict__ gmax,
                     const float* __restrict__ gsum, float* __restrict__ xg, int N) {
    __shared__ float As[64 * D];                 // 64 KB, swizzled
    const int t = threadIdx.x;
    const int m0 = blockIdx.x * 64;

    // cooperative load: 64x256 fp32 as float4, coalesced (global_load_b128)
#pragma unroll
    for (int i = 0; i < 16; ++i) {
        int f = i * 256 + t;                     // float4 index
        int row = f >> 6;                        // 64 float4 per row
        int col = (f & 63) * 4;
        float4 v = make_float4(0.f, 0.f, 0.f, 0.f);
        if (m0 + row < N)
            v = *(const float4*)(x + (size_t)(m0 + row) * D + col);
        int sc = col ^ ((row & 15) << 2);        // bank-conflict-free swizzle
        float* dst = &As[row * D + sc];
        dst[0] = v.x; dst[1] = v.y; dst[2] = v.z; dst[3] = v.w;
    }
    __syncthreads();

    const int wave = t >> 5, lane = t & 31;
    const int n0 = wave * 32;
    const int nh = lane & 15;                    // N within 16 / M row for A frag
    const int kh = (lane >> 4) * 2;              // K sub-offset 0 or 2

    v8f acc[4][2] = {};
    for (int k0 = 0; k0 < D; k0 += 4) {
        int krow = k0 + kh;
        v2f b0, b1;                              // B frags for two 16-col tiles
        b0.x = Wf[(size_t)krow * D + n0 + nh];
        b0.y = Wf[(size_t)(krow + 1) * D + n0 + nh];
        b1.x = Wf[(size_t)krow * D + n0 + 16 + nh];
        b1.y = Wf[(size_t)(krow + 1) * D + n0 + 16 + nh];
#pragma unroll
        for (int ms = 0; ms < 4; ++ms) {
            const float* ap = &As[(ms * 16 + nh) * D + (krow ^ (nh << 2))];
            v2f a; a.x = ap[0]; a.y = ap[1];     // ds_load_b64, conflict-free
            acc[ms][0] = wmma_f32(a, b0, acc[ms][0]);
            acc[ms][1] = wmma_f32(a, b1, acc[ms][1]);
        }
    }

    // epilogue: bias + LeakyReLU + alpha scaling + atomic segment accumulate
#pragma unroll
    for (int ms = 0; ms < 4; ++ms) {
#pragma unroll
        for (int nt = 0; nt < 2; ++nt) {
            int ncol = n0 + nt * 16 + nh;
            float bias = bf[ncol];
#pragma unroll
            for (int r = 0; r < 8; ++r) {
                int row = m0 + ms * 16 + r + ((lane >> 4) * 8);
                if (row < N) {
                    float v = leaky(acc[ms][nt][r] + bias);
                    int b = batch[row];
                    float alpha = __expf(gate[row] - gmax[b]) / gsum[b];
                    atomicAdd(&xg[(size_t)b * D + ncol], alpha * v);
                }
            }
        }
    }
}

// ------------- out = leaky([xg | xg_old] @ W_t + b_t) + xg_old  (WMMA)
// block: 16 graphs x 256 cols, K = 512; B=10000 -> 625 blocks exactly
__global__ __launch_bounds__(256)
void transform_kernel(const float* __restrict__ xg, const float* __restrict__ xg_old,
                      const float* __restrict__ Wt, const float* __restrict__ bt,
                      float* __restrict__ out) {
    __shared__ float As[16 * 512];               // 32 KB, swizzled
    const int t = threadIdx.x;
    const int g0 = blockIdx.x * 16;

#pragma unroll
    for (int i = 0; i < 8; ++i) {
        int f = i * 256 + t;                     // float4 index
        int row = f >> 7;                        // 128 float4 per row
        int col = (f & 127) * 4;
        const float* src = (col < D)
            ? (xg + (size_t)(g0 + row) * D + col)
            : (xg_old + (size_t)(g0 + row) * D + (col - D));
        float4 v = *(const float4*)src;
        int sc = col ^ ((row & 15) << 2);
        float* dst = &As[row * 512 + sc];
        dst[0] = v.x; dst[1] = v.y; dst[2] = v.z; dst[3] = v.w;
    }
    __syncthreads();

    const int wave = t >> 5, lane = t & 31;
    const int n0 = wave * 32;
    const int nh = lane & 15;
    const int kh = (lane >> 4) * 2;

    v8f acc0 = {}, acc1 = {};
    for (int k0 = 0; k0 < 512; k0 += 4) {
        int krow = k0 + kh;
        v2f b0, b1;
        b0.x = Wt[(size_t)krow * D + n0 + nh];
        b0.y = Wt[(size_t)(krow + 1) * D + n0 + nh];
        b1.x = Wt[(size_t)krow * D + n0 + 16 + nh];
        b1.y = Wt[(size_t)(krow + 1) * D + n0 + 16 + nh];
        const float* ap = &As[nh * 512 + (krow ^ (nh << 2))];
        v2f a; a.x = ap[0]; a.y = ap[1];
        acc0 = wmma_f32(a, b0, acc0);
        acc1 = wmma_f32(a, b1, acc1);
    }

#pragma unroll
    for (int nt = 0; nt < 2; ++nt) {
        v8f c = nt ? acc1 : acc0;
        int ncol = n0 + nt * 16 + nh;
        float bias = bt[ncol];
#pragma unroll
        for (int r = 0; r < 8; ++r) {
            int row = g0 + r + ((lane >> 4) * 8);
            float old = xg_old[(size_t)row * D + ncol];
            out[(size_t)row * D + ncol] = leaky(c[r] + bias) + old;
        }
    }
}

// ----------------------------------------------------------------------
extern "C" void kernel_launch(void* const* d_in, const int* in_sizes, int n_in,
                              void* d_out, int out_size, void* d_ws, size_t ws_size,
                              hipStream_t stream) {
    const float* xg_old = (const float*)d_in[0];
    const float* x      = (const float*)d_in[1];
    const int*   batch  = (const int*)d_in[2];
    const float* W_gate = (const float*)d_in[3];
    const float* b_gate = (const float*)d_in[4];
    const float* W_feat = (const float*)d_in[5];
    const float* b_feat = (const float*)d_in[6];
    const float* W_t    = (const float*)d_in[7];
    const float* b_t    = (const float*)d_in[8];
    float* out = (float*)d_out;

    const int N = in_sizes[2];           // 500000
    const int B = in_sizes[0] / D;       // 10000

    // workspace layout (fp32): gate[N] | gmax[B] | gsum[B] | xg[B*D]
    float* gate = (float*)d_ws;
    float* gmax = gate + N;
    float* gsum = gmax + B;
    float* xg   = gsum + B;

    init_kernel<<<(B * D + 255) / 256, 256, 0, stream>>>(gmax, gsum, xg, B, B * D);
    gate_max_kernel<<<(N + 7) / 8, 256, 0, stream>>>(x, batch, W_gate, b_gate,
                                                     gate, gmax, N);
    expsum_kernel<<<(N + 255) / 256, 256, 0, stream>>>(gate, batch, gmax, gsum, N);
    feat_agg_kernel<<<(N + 63) / 64, 256, 0, stream>>>(x, batch, W_feat, b_feat,
                                                       gate, gmax, gsum, xg, N);
    transform_kernel<<<B / 16, 256, 0, stream>>>(xg, xg_old, W_t, b_t, out);
}